// QubitClassifier_66975720014174
// MI455X (gfx1250) — compile-verified
//
#include <hip/hip_runtime.h>
#include <math.h>

#define NQ 13
#define DIM 8192
#define NLAYERS 5
#define BLOCK 256
#define PI_F 3.14159265358979323846f

// dynamic LDS layout (floats):
//   re[8192] im[8192] ure[3*256] uim[3*256] uimn[3*256] h16[256]
//   gre[13*4] gim[13*4] xv[16] red[256]
#define SMEM_FLOATS (DIM*2 + 768*3 + 256 + 52 + 52 + 16 + 256)

typedef float v2f __attribute__((ext_vector_type(2)));
typedef float v8f __attribute__((ext_vector_type(8)));

__device__ __forceinline__ v8f wmma_k4(v2f a, v2f b, v8f c) {
  // D(16x16,f32) += A(16x4,f32) * B(4x16,f32)
  return __builtin_amdgcn_wmma_f32_16x16x4_f32(false, a, false, b, (short)0, c,
                                               false, false);
}

// Apply real 16x16 matrix A (row-major [M*16+K]) to the 4-bit index group whose
// low bit is p, on both re and im planes. State index: s = a*(16<<p) + r*(1<<p) + b.
__device__ __forceinline__ void apply16_real(float* __restrict__ re,
                                             float* __restrict__ im,
                                             const float* __restrict__ A,
                                             int p, int tid) {
  const int lane = tid & 31, wave = tid >> 5;
  const int n = lane & 15, hi = lane >> 4;
  const int pm = (1 << p) - 1;
#pragma unroll
  for (int t = 0; t < 4; ++t) {
    const int c = (wave * 4 + t) * 16 + n;                 // column 0..511
    const int sb = ((c >> p) << (p + 4)) + (c & pm);       // base state index
    v8f accR = {0.f, 0.f, 0.f, 0.f, 0.f, 0.f, 0.f, 0.f};
    v8f accI = {0.f, 0.f, 0.f, 0.f, 0.f, 0.f, 0.f, 0.f};
#pragma unroll
    for (int k0 = 0; k0 < 16; k0 += 4) {
      const int k = k0 + hi * 2;                           // K index for this half-wave
      v2f a  = { A[n * 16 + k], A[n * 16 + k + 1] };
      v2f br = { re[sb + (k << p)], re[sb + ((k + 1) << p)] };
      v2f bi = { im[sb + (k << p)], im[sb + ((k + 1) << p)] };
      accR = wmma_k4(a, br, accR);
      accI = wmma_k4(a, bi, accI);
    }
#pragma unroll
    for (int v = 0; v < 8; ++v) {
      const int r = v + hi * 8;
      re[sb + (r << p)] = accR[v];
      im[sb + (r << p)] = accI[v];
    }
  }
}

// Complex 16x16 matrix (Are + i*Aim, with Ain = -Aim precomputed) applied to
// the complex state on index group with low bit p.
__device__ __forceinline__ void apply16_cplx(float* __restrict__ re,
                                             float* __restrict__ im,
                                             const float* __restrict__ Are,
                                             const float* __restrict__ Aim,
                                             const float* __restrict__ Ain,
                                             int p, int tid) {
  const int lane = tid & 31, wave = tid >> 5;
  const int n = lane & 15, hi = lane >> 4;
  const int pm = (1 << p) - 1;
#pragma unroll
  for (int t = 0; t < 4; ++t) {
    const int c = (wave * 4 + t) * 16 + n;
    const int sb = ((c >> p) << (p + 4)) + (c & pm);
    v8f accR = {0.f, 0.f, 0.f, 0.f, 0.f, 0.f, 0.f, 0.f};
    v8f accI = {0.f, 0.f, 0.f, 0.f, 0.f, 0.f, 0.f, 0.f};
#pragma unroll
    for (int k0 = 0; k0 < 16; k0 += 4) {
      const int k = k0 + hi * 2;
      v2f ar = { Are[n * 16 + k], Are[n * 16 + k + 1] };
      v2f ai = { Aim[n * 16 + k], Aim[n * 16 + k + 1] };
      v2f an = { Ain[n * 16 + k], Ain[n * 16 + k + 1] };
      v2f br = { re[sb + (k << p)], re[sb + ((k + 1) << p)] };
      v2f bi = { im[sb + (k << p)], im[sb + ((k + 1) << p)] };
      accR = wmma_k4(ar, br, accR);   // + Ar*Br
      accR = wmma_k4(an, bi, accR);   // - Ai*Bi
      accI = wmma_k4(ar, bi, accI);   // + Ar*Bi
      accI = wmma_k4(ai, br, accI);   // + Ai*Br
    }
#pragma unroll
    for (int v = 0; v < 8; ++v) {
      const int r = v + hi * 8;
      re[sb + (r << p)] = accR[v];
      im[sb + (r << p)] = accI[v];
    }
  }
}

__global__ void __launch_bounds__(BLOCK)
qubit_classifier_kernel(const float* __restrict__ x,
                        const float* __restrict__ theta,
                        const float* __restrict__ bias,
                        float* __restrict__ out) {
  extern __shared__ float smem[];
  float* re   = smem;
  float* im   = re + DIM;
  float* ure  = im + DIM;         // 3*256
  float* uim  = ure + 768;        // 3*256
  float* uimn = uim + 768;        // 3*256
  float* h16  = uimn + 768;       // 256
  float* gre  = h16 + 256;        // 13*4 per-qubit gate entries (re)
  float* gim  = gre + 52;         // 13*4 (im)
  float* xv   = gim + 52;         // 16
  float* red  = xv + 16;          // 256

  const int tid = threadIdx.x;
  const int b = blockIdx.x;

  // H^{x4} 16x16 with folded 1/sqrt(16) normalization: entries +-0.25
  h16[tid] = (__popc((tid >> 4) & (tid & 15)) & 1) ? -0.25f : 0.25f;
  if (tid < NQ) xv[tid] = x[b * NQ + tid];
  __syncthreads();

  float aa = 0.f;
#pragma unroll
  for (int j = 0; j < NQ; ++j) { float a = PI_F - xv[j]; aa += a * a; }

  // ---- feature map: psi = d * FWHT(d * |uniform>) ----
  const float c0 = 0.011048543456039806f;   // DIM^{-1/2} = 2^{-6.5}
  for (int s = tid; s < DIM; s += BLOCK) {
    float u = 0.f, v = 0.f;
#pragma unroll
    for (int j = 0; j < NQ; ++j) {
      float sg = ((s >> (12 - j)) & 1) ? -1.f : 1.f;  // qubit j <-> bit 12-j
      u += xv[j] * sg;
      v += (PI_F - xv[j]) * sg;
    }
    float phi = u + 0.5f * (v * v - aa);
    float sp, cp; __sincosf(phi, &sp, &cp);
    re[s] = c0 * cp; im[s] = c0 * sp;
  }
  __syncthreads();

  // normalized FWHT = three 16x16 Hadamard matmuls + one sqrt(1/2) butterfly
  apply16_real(re, im, h16, 9, tid); __syncthreads();
  apply16_real(re, im, h16, 5, tid); __syncthreads();
  apply16_real(re, im, h16, 1, tid); __syncthreads();
  const float rs2 = 0.70710678118654752f;
  for (int pr = tid; pr < DIM / 2; pr += BLOCK) {
    int s0 = pr * 2;
    float r0 = re[s0], i0 = im[s0], r1 = re[s0 + 1], i1 = im[s0 + 1];
    re[s0]     = rs2 * (r0 + r1); im[s0]     = rs2 * (i0 + i1);
    re[s0 + 1] = rs2 * (r0 - r1); im[s0 + 1] = rs2 * (i0 - i1);
  }
  __syncthreads();

  // second diagonal phase
  for (int s = tid; s < DIM; s += BLOCK) {
    float u = 0.f, v = 0.f;
#pragma unroll
    for (int j = 0; j < NQ; ++j) {
      float sg = ((s >> (12 - j)) & 1) ? -1.f : 1.f;
      u += xv[j] * sg;
      v += (PI_F - xv[j]) * sg;
    }
    float phi = u + 0.5f * (v * v - aa);
    float sp, cp; __sincosf(phi, &sp, &cp);
    float r = re[s], i2 = im[s];
    re[s] = r * cp - i2 * sp;
    im[s] = r * sp + i2 * cp;
  }
  __syncthreads();

  // ---- variational layers ----
  for (int l = 0; l < NLAYERS; ++l) {
    if (l > 0) {  // CZ-chain diagonal: sign = (-1)^popcount(s & (s>>1))
      for (int s = tid; s < DIM; s += BLOCK) {
        if (__popc(s & (s >> 1)) & 1) { re[s] = -re[s]; im[s] = -im[s]; }
      }
    }
    if (tid < NQ) {   // per-qubit 2x2 gate entries Rz(b)Ry(a)
      float al = theta[(l * NQ + tid) * 2 + 0];
      float be = theta[(l * NQ + tid) * 2 + 1];
      float sa, ca, sb2, cb;
      __sincosf(0.5f * al, &sa, &ca);
      __sincosf(0.5f * be, &sb2, &cb);
      gre[tid * 4 + 0] =  cb * ca;  gim[tid * 4 + 0] = -sb2 * ca;  // em*ca
      gre[tid * 4 + 1] = -cb * sa;  gim[tid * 4 + 1] =  sb2 * sa;  // -em*sa
      gre[tid * 4 + 2] =  cb * sa;  gim[tid * 4 + 2] =  sb2 * sa;  // ep*sa
      gre[tid * 4 + 3] =  cb * ca;  gim[tid * 4 + 3] =  sb2 * ca;  // ep*ca
    }
    __syncthreads();
    {  // build 3 Kronecker 16x16 group matrices (qubits 4g..4g+3)
      const int r = tid >> 4, c = tid & 15;
#pragma unroll
      for (int g = 0; g < 3; ++g) {
        float pr = 1.f, pi = 0.f;
#pragma unroll
        for (int j = 0; j < 4; ++j) {
          int q = g * 4 + j;                        // qubit q <-> group bit 3-j
          int rj = (r >> (3 - j)) & 1, cj = (c >> (3 - j)) & 1;
          float er = gre[q * 4 + rj * 2 + cj], ei = gim[q * 4 + rj * 2 + cj];
          float nr = pr * er - pi * ei;
          float ni = pr * ei + pi * er;
          pr = nr; pi = ni;
        }
        ure[g * 256 + tid] = pr;
        uim[g * 256 + tid] = pi;
        uimn[g * 256 + tid] = -pi;
      }
    }
    __syncthreads();
    apply16_cplx(re, im, ure,       uim,       uimn,       9, tid); __syncthreads();
    apply16_cplx(re, im, ure + 256, uim + 256, uimn + 256, 5, tid); __syncthreads();
    apply16_cplx(re, im, ure + 512, uim + 512, uimn + 512, 1, tid); __syncthreads();
    {  // qubit 12 (bit 0) conventional 2x2 gate
      const float u00r = gre[48], u00i = gim[48];
      const float u01r = gre[49], u01i = gim[49];
      const float u10r = gre[50], u10i = gim[50];
      const float u11r = gre[51], u11i = gim[51];
      for (int pr2 = tid; pr2 < DIM / 2; pr2 += BLOCK) {
        int s0 = pr2 * 2;
        float r0 = re[s0], i0 = im[s0], r1 = re[s0 + 1], i1 = im[s0 + 1];
        re[s0]     = u00r * r0 - u00i * i0 + u01r * r1 - u01i * i1;
        im[s0]     = u00r * i0 + u00i * r0 + u01r * i1 + u01i * r1;
        re[s0 + 1] = u10r * r0 - u10i * i0 + u11r * r1 - u11i * i1;
        im[s0 + 1] = u10r * i0 + u10i * r0 + u11r * i1 + u11i * r1;
      }
    }
    __syncthreads();
  }

  // ---- parity-weighted probability readout ----
  float part = 0.f;
  for (int s = tid; s < DIM; s += BLOCK) {
    float m = re[s] * re[s] + im[s] * im[s];
    part += (__popc(s) & 1) ? -m : m;
  }
  red[tid] = part;
  __syncthreads();
  for (int off = BLOCK / 2; off > 0; off >>= 1) {
    if (tid < off) red[tid] += red[tid + off];
    __syncthreads();
  }
  if (tid == 0) {
    float logit = red[0] + bias[0];
    out[b * 2 + 0] = -logit;
    out[b * 2 + 1] = logit;
  }
}

extern "C" void kernel_launch(void* const* d_in, const int* in_sizes, int n_in,
                              void* d_out, int out_size, void* d_ws, size_t ws_size,
                              hipStream_t stream) {
  const float* x     = (const float*)d_in[0];
  const float* theta = (const float*)d_in[1];
  const float* bias  = (const float*)d_in[2];
  float* out = (float*)d_out;
  const int batch = in_sizes[0] / NQ;   // 512
  const size_t shmem = (size_t)SMEM_FLOATS * sizeof(float);  // ~76 KB < 320 KB/WGP
  qubit_classifier_kernel<<<batch, BLOCK, shmem, stream>>>(x, theta, bias, out);
}